// ImportanceEncoder_5214090297373
// MI455X (gfx1250) — compile-verified
//
#include <hip/hip_runtime.h>

typedef __attribute__((ext_vector_type(2))) float v2f;
typedef __attribute__((ext_vector_type(4))) float v4f;
typedef __attribute__((ext_vector_type(8))) float v8f;

#define BB 8
#define SS 43
#define NQ 13
#define KK 4
#define DD 608

__global__ __launch_bounds__(32)
void importance_encoder_wmma(const float* __restrict__ x,
                             const float* __restrict__ scores,
                             const float* __restrict__ Wff,
                             const float* __restrict__ bff,
                             float* __restrict__ out)
{
    __shared__ int   idx_lds[NQ][KK];
    __shared__ float c_lds[16][16];

    const int b    = blockIdx.x;
    const int lane = threadIdx.x;    // 0..31, wave32
    const int half = lane >> 4;      // 0 or 1
    const int row  = lane & 15;      // M index for A, N index for B/C

    const float* xb = x + (size_t)b * SS * DD;

    // ---- Phase 1: top-K of scores[b, i, :NQ]; lane i handles query i ----
    if (lane < NQ) {
        float v[NQ];
        const float* srow = scores + ((size_t)b * SS + lane) * SS;
        #pragma unroll
        for (int q = 0; q < NQ; ++q) v[q] = srow[q];
        unsigned used = 0u;
        #pragma unroll
        for (int k = 0; k < KK; ++k) {
            float best = -__builtin_inff();
            int   bi   = 0;
            #pragma unroll
            for (int q = 0; q < NQ; ++q) {
                // strict '>' => ties keep the lower index (matches lax.top_k)
                bool better = (((used >> q) & 1u) == 0u) && (v[q] > best);
                best = better ? v[q] : best;
                bi   = better ? q    : bi;
            }
            used |= 1u << bi;
            idx_lds[lane][k] = bi;
        }
    }

    // ---- Phase 2: C[16x16] = Xpad[16x608] @ Wpad[608x16] via v_wmma_f32_16x16x4_f32
    // Columns 0..3 of Wpad = W_ff[:608, j] (self half), 4..7 = W_ff[608:, j] (gather half).
    const float aScale = (row < NQ) ? 1.0f : 0.0f;          // zero-pad rows 13..15
    const float* arow  = xb + (size_t)((row < NQ) ? row : 0) * DD;

    float wScale = 1.0f;                                    // zero-pad cols 8..15
    const float* wbase;
    if (row < 4)      { wbase = Wff + row; }                // W1 column `row`
    else if (row < 8) { wbase = Wff + DD * 4 + (row - 4); } // W2 column `row-4`
    else              { wbase = Wff; wScale = 0.0f; }

    v8f c = {0.f, 0.f, 0.f, 0.f, 0.f, 0.f, 0.f, 0.f};
    #pragma unroll 4
    for (int k0 = 0; k0 < DD; k0 += 4) {
        const int k = k0 + 2 * half;     // this lane's K pair (same for A and B)
        v2f a = *(const v2f*)(arow + k); // 8B-aligned: base 16B, row stride 2432B, k even
        a.x *= aScale;
        a.y *= aScale;
        v2f bv;
        bv.x = wbase[4 * k]     * wScale;  // W row k,   this lane's column
        bv.y = wbase[4 * k + 4] * wScale;  // W row k+1
        c = __builtin_amdgcn_wmma_f32_16x16x4_f32(
                false, a, false, bv, (short)0, c, false, false);
    }

    // Spill C to LDS: c[v] in lane L -> row v + 8*(L>>4), col L&15
    #pragma unroll
    for (int v = 0; v < 8; ++v)
        c_lds[v + 8 * half][row] = c[v];

    __syncthreads();

    // ---- Phase 3: write out[b,i,q,:] with no races (decide each cell directly) ----
    const float b0 = bff[0], b1 = bff[1], b2 = bff[2], b3 = bff[3];
    float* outb = out + (size_t)b * NQ * NQ * 4;
    for (int cell = lane; cell < NQ * NQ; cell += 32) {
        const int i = cell / NQ;
        const int q = cell - i * NQ;
        int hit = -1;
        #pragma unroll
        for (int k = 0; k < KK; ++k)
            if (idx_lds[i][k] == q) hit = k;   // idx entries are distinct per (b,i)
        v4f o;
        if (hit >= 0) {
            o.x = c_lds[i][0] + c_lds[q][4] + b0;
            o.y = c_lds[i][1] + c_lds[q][5] + b1;
            o.z = c_lds[i][2] + c_lds[q][6] + b2;
            o.w = c_lds[i][3] + c_lds[q][7] + b3;
        } else {
            o = (v4f){1.f, 0.f, 0.f, 0.f};
        }
        *(v4f*)(outb + cell * 4) = o;          // 16B aligned: cell*16 bytes
    }
}

extern "C" void kernel_launch(void* const* d_in, const int* in_sizes, int n_in,
                              void* d_out, int out_size, void* d_ws, size_t ws_size,
                              hipStream_t stream) {
    const float* x      = (const float*)d_in[0];
    const float* scores = (const float*)d_in[1];
    const float* Wff    = (const float*)d_in[2];
    const float* bff    = (const float*)d_in[3];
    float* out          = (float*)d_out;
    importance_encoder_wmma<<<dim3(BB), dim3(32), 0, stream>>>(x, scores, Wff, bff, out);
}